// FastRCNNOutputLayers_baseline_23794118820562
// MI455X (gfx1250) — compile-verified
//
#include <hip/hip_runtime.h>
#include <stdint.h>

#define B_IMG 16
#define R_ROWS 4096
#define K_CLS 80
#define SCORE_COLS 81
#define BOX_COLS 320
#define CAP 16384
#define NPRE 1024
#define TOPK_OUT 100
#define IMG_W_F 1333.0f
#define IMG_H_F 800.0f
#define SCORE_TH 0.05f
#define MAXC 2000.0f

typedef float v2f __attribute__((ext_vector_type(2)));
typedef float v8f __attribute__((ext_vector_type(8)));

#if __has_builtin(__builtin_amdgcn_wmma_f32_16x16x4_f32)
#define HAVE_WMMA4 1
#else
#define HAVE_WMMA4 0
#endif

// ---------------- kernel 0: zero per-image candidate counters ----------------
__global__ void k_init(int* counts) {
    if (threadIdx.x < B_IMG) counts[threadIdx.x] = 0;
}

// ---------------- kernel 1: per-row validity (all 320 box + 81 score finite) -
__global__ void k_valid(const float* __restrict__ boxes,
                        const float* __restrict__ scores,
                        unsigned char* __restrict__ rowValid) {
    int wave = (int)((blockIdx.x * blockDim.x + threadIdx.x) >> 5);
    int lane = threadIdx.x & 31;
    if (wave >= B_IMG * R_ROWS) return;
    const float* brow = boxes + (size_t)wave * BOX_COLS;
    const float* srow = scores + (size_t)wave * SCORE_COLS;
    bool bad = false;
    for (int i = lane; i < BOX_COLS; i += 32) bad |= !__builtin_isfinite(brow[i]);
    for (int i = lane; i < SCORE_COLS; i += 32) bad |= !__builtin_isfinite(srow[i]);
    unsigned anybad = __builtin_amdgcn_ballot_w32(bad);
    if (lane == 0) rowValid[wave] = (anybad == 0u) ? 1 : 0;
}

// ---------------- kernel 2: streaming filter -> packed (score,~idx) keys -----
__global__ void k_gather(const float* __restrict__ scores,
                         const unsigned char* __restrict__ rowValid,
                         int* __restrict__ counts,
                         unsigned long long* __restrict__ cand) {
    long g = (long)blockIdx.x * blockDim.x + threadIdx.x;
    const long total = (long)B_IMG * R_ROWS * K_CLS;
    if (g >= total) return;
    int b = (int)(g / ((long)R_ROWS * K_CLS));
    int rem = (int)(g - (long)b * R_ROWS * K_CLS);
    int r = rem / K_CLS, c = rem % K_CLS;
    float s = scores[((size_t)b * R_ROWS + r) * SCORE_COLS + c];
    if (s > SCORE_TH && rowValid[b * R_ROWS + r]) {
        int pos = atomicAdd(&counts[b], 1);
        if (pos < CAP) {
            unsigned idx = (unsigned)(r * K_CLS + c);
            unsigned long long key =
                ((unsigned long long)__float_as_uint(s) << 32) | (unsigned)(~idx);
            cand[(size_t)b * CAP + pos] = key;
        }
    }
}

// ---------------- kernel 3: in-LDS bitonic sort (desc) + top-1024 select -----
__global__ __launch_bounds__(1024) void k_sortsel(
        const float* __restrict__ boxes,
        const unsigned long long* __restrict__ cand,
        const int* __restrict__ counts,
        float* __restrict__ selScore,
        float* __restrict__ selBox,
        int* __restrict__ selCls) {
    __shared__ unsigned long long keys[CAP];   // 128 KB LDS working set
    const int b = blockIdx.x;
    const int tid = threadIdx.x;
    int c = counts[b];
    if (c > CAP) c = CAP;
    const unsigned long long* src = cand + (size_t)b * CAP;

    // Stage candidates into LDS via CDNA5 async global->LDS copies.
    for (int i = tid; i < CAP; i += 1024) {
        if (i < c) {
            unsigned ldsoff = (unsigned)(uintptr_t)(&keys[i]);
            unsigned long long ga = (unsigned long long)(uintptr_t)(src + i);
            asm volatile("global_load_async_to_lds_b64 %0, %1, off"
                         :: "v"(ldsoff), "v"(ga) : "memory");
        } else {
            keys[i] = 0ull;   // pad: sorts below every real key
        }
    }
    asm volatile("s_wait_asynccnt 0" ::: "memory");
    __syncthreads();

    // Bitonic sort, descending (key = score_bits:~idx matches top_k tiebreak).
    for (unsigned k = 2; k <= CAP; k <<= 1) {
        for (unsigned j = k >> 1; j > 0; j >>= 1) {
            for (unsigned i = tid; i < CAP; i += 1024) {
                unsigned p = i ^ j;
                if (p > i) {
                    unsigned long long x = keys[i], y = keys[p];
                    bool doswap = ((i & k) == 0) ? (x < y) : (x > y);
                    if (doswap) { keys[i] = y; keys[p] = x; }
                }
            }
            __syncthreads();
        }
    }

    // Emit slot tid of the top-1024.
    unsigned long long key = keys[tid];
    size_t o = (size_t)b * NPRE + tid;
    float* bo = selBox + o * 4;
    if (key == 0ull) {
        selScore[o] = -__builtin_inff();
        selCls[o] = 0;
        bo[0] = bo[1] = bo[2] = bo[3] = 0.f;
    } else {
        float s = __uint_as_float((unsigned)(key >> 32));
        unsigned idx = ~((unsigned)key);
        int r = (int)(idx / K_CLS), cl = (int)(idx % K_CLS);
        const float* bp = boxes + ((size_t)b * R_ROWS + r) * BOX_COLS + (size_t)cl * 4;
        float x1 = fminf(fmaxf(bp[0], 0.f), IMG_W_F);
        float y1 = fminf(fmaxf(bp[1], 0.f), IMG_H_F);
        float x2 = fminf(fmaxf(bp[2], 0.f), IMG_W_F);
        float y2 = fminf(fmaxf(bp[3], 0.f), IMG_H_F);
        selScore[o] = s;
        selCls[o] = cl;
        bo[0] = x1; bo[1] = y1; bo[2] = x2; bo[3] = y2;
    }
}

// ---------------- kernel 4: WMMA-tiled IoU bitmatrix + greedy NMS + top-100 --
__global__ __launch_bounds__(1024) void k_nms(
        const float* __restrict__ selScore,
        const float* __restrict__ selBox,
        const int* __restrict__ selCls,
        float* __restrict__ outAll) {
    __shared__ unsigned mask[NPRE * 32];   // 128 KB: 1024x1024 suppression bits
    __shared__ float4 obox[NPRE];          // class-offset boxes
    __shared__ float areaS[NPRE];
    __shared__ unsigned char keep[NPRE];
    __shared__ unsigned keepW[32];
    __shared__ float stS[TOPK_OUT];
    __shared__ int stC[TOPK_OUT];
    __shared__ float stB[TOPK_OUT * 4];

    const int b = blockIdx.x, tid = threadIdx.x;
    const int lane = tid & 31, wv = tid >> 5;

    {   // load phase: offset boxes + areas + keep0 = isfinite(score)
        size_t o = (size_t)b * NPRE + tid;
        float sc = selScore[o];
        const float* bp = selBox + o * 4;
        float off = (float)selCls[o] * MAXC;
        float x1 = bp[0] + off, y1 = bp[1] + off, x2 = bp[2] + off, y2 = bp[3] + off;
        obox[tid] = make_float4(x1, y1, x2, y2);
        areaS[tid] = (x2 - x1) * (y2 - y1);
        keep[tid] = (sc > -1e30f) ? 1 : 0;
    }
    for (int w = 0; w < 32; ++w) mask[tid * 32 + w] = 0u;
    __syncthreads();

    // 64x64 grid of 16x16 tiles; one tile per wave iteration.
    const bool lo = lane < 16;
    const int l15 = lane & 15;
    for (int t = wv; t < 64 * 64; t += 32) {
        int i16 = (t >> 6) << 4;
        int tj = t & 63;
        int j16 = tj << 4;
#if HAVE_WMMA4
        // area_i + area_j as a rank-2 product: A(16x4)=[area|1|0|0], B(4x16)=[1;area;0;0]
        v2f a, bb;
        a.x = lo ? areaS[i16 + l15] : 0.f;     // K=0 (lanes 0-15) / K=2 zeros
        a.y = lo ? 1.0f : 0.f;                 // K=1 / K=3 zeros
        bb.x = lo ? 1.0f : 0.f;                // row K=0 ones / K=2 zeros
        bb.y = lo ? areaS[j16 + l15] : 0.f;    // row K=1 areas / K=3 zeros
        v8f cacc = {0.f, 0.f, 0.f, 0.f, 0.f, 0.f, 0.f, 0.f};
        v8f d = __builtin_amdgcn_wmma_f32_16x16x4_f32(
            false, a, false, bb, (short)0, cacc, false, false);
#endif
        float4 bj = obox[j16 + l15];
        int mbase = lo ? 0 : 8;
#pragma unroll
        for (int r = 0; r < 8; ++r) {
            int i = i16 + mbase + r;
            float4 bi = obox[i];
            float lx = fmaxf(bi.x, bj.x), ly = fmaxf(bi.y, bj.y);
            float rx = fminf(bi.z, bj.z), ry = fminf(bi.w, bj.w);
            float iw = fmaxf(rx - lx, 0.f), ih = fmaxf(ry - ly, 0.f);
            float inter = iw * ih;
#if HAVE_WMMA4
            float asum = d[r];
#else
            float asum = areaS[i] + areaS[j16 + l15];
#endif
            // iou > 0.5  <=>  inter/(asum-inter) > 0.5  <=>  3*inter > asum
            bool cond = (3.0f * inter > asum);
            unsigned bal = __builtin_amdgcn_ballot_w32(cond);
            if (lane == 0) {
                unsigned wsel = (unsigned)(tj >> 1);
                unsigned sh = (unsigned)((tj & 1) * 16);
                atomicOr(&mask[(i16 + r) * 32 + wsel], (bal & 0xFFFFu) << sh);
                atomicOr(&mask[(i16 + 8 + r) * 32 + wsel], (bal >> 16) << sh);
            }
        }
    }
    __syncthreads();

    // Greedy sequential NMS over sorted candidates (bitmatrix rows in LDS).
    for (int i = 0; i < NPRE; ++i) {
        if (keep[i] && tid > i) {
            if ((mask[i * 32 + (tid >> 5)] >> (tid & 31)) & 1u) keep[tid] = 0;
        }
        __syncthreads();
    }

    // Compact: scores already sorted desc, so top-100 = first 100 survivors.
    if (tid < 32) {
        unsigned w = 0;
        for (int k2 = 0; k2 < 32; ++k2)
            w |= ((unsigned)keep[tid * 32 + k2]) << k2;
        keepW[tid] = w;
    }
    if (tid < TOPK_OUT) {
        stS[tid] = 0.f;
        stC[tid] = -1;
        stB[tid * 4 + 0] = stB[tid * 4 + 1] = stB[tid * 4 + 2] = stB[tid * 4 + 3] = 0.f;
    }
    __syncthreads();
    if (keep[tid]) {
        int ws = tid >> 5;
        int rank = __popc(keepW[ws] & ((1u << (tid & 31)) - 1u));
        for (int k2 = 0; k2 < ws; ++k2) rank += __popc(keepW[k2]);
        if (rank < TOPK_OUT) {
            size_t o = (size_t)b * NPRE + tid;
            stS[rank] = selScore[o];
            stC[rank] = selCls[o];
            const float* bp = selBox + o * 4;  // un-offset clipped box
            stB[rank * 4 + 0] = bp[0];
            stB[rank * 4 + 1] = bp[1];
            stB[rank * 4 + 2] = bp[2];
            stB[rank * 4 + 3] = bp[3];
        }
    }
    __syncthreads();
    if (tid < TOPK_OUT) {
        float* outB = outAll;                                  // 16*100*4 floats
        float* outS = outAll + B_IMG * TOPK_OUT * 4;           // 16*100 floats
        int* outC = (int*)outAll + B_IMG * TOPK_OUT * 5;       // 16*100 ints
        size_t ob = (size_t)b * TOPK_OUT + tid;
        outB[ob * 4 + 0] = stB[tid * 4 + 0];
        outB[ob * 4 + 1] = stB[tid * 4 + 1];
        outB[ob * 4 + 2] = stB[tid * 4 + 2];
        outB[ob * 4 + 3] = stB[tid * 4 + 3];
        outS[ob] = stS[tid];
        outC[ob] = stC[tid];
    }
}

// ---------------- launcher ---------------------------------------------------
extern "C" void kernel_launch(void* const* d_in, const int* in_sizes, int n_in,
                              void* d_out, int out_size, void* d_ws, size_t ws_size,
                              hipStream_t stream) {
    const float* boxes = (const float*)d_in[0];
    const float* scores = (const float*)d_in[1];

    char* w = (char*)d_ws;
    unsigned char* rowValid = (unsigned char*)w;                         // 65536 B
    int* counts = (int*)(w + 65536);                                     // 64 B
    unsigned long long* cand = (unsigned long long*)(w + 65600);         // 2 MB
    float* selScore = (float*)(w + 65600 + 2097152);                     // 64 KB
    float* selBox = (float*)(w + 65600 + 2097152 + 65536);               // 256 KB
    int* selCls = (int*)(w + 65600 + 2097152 + 65536 + 262144);          // 64 KB

    hipLaunchKernelGGL(k_init, dim3(1), dim3(64), 0, stream, counts);
    hipLaunchKernelGGL(k_valid, dim3((B_IMG * R_ROWS) / 8), dim3(256), 0, stream,
                       boxes, scores, rowValid);
    long total = (long)B_IMG * R_ROWS * K_CLS;
    hipLaunchKernelGGL(k_gather, dim3((unsigned)((total + 255) / 256)), dim3(256), 0,
                       stream, scores, rowValid, counts, cand);
    hipLaunchKernelGGL(k_sortsel, dim3(B_IMG), dim3(1024), 0, stream,
                       boxes, cand, counts, selScore, selBox, selCls);
    hipLaunchKernelGGL(k_nms, dim3(B_IMG), dim3(1024), 0, stream,
                       selScore, selBox, selCls, (float*)d_out);
}